// ShiftNMF_23691039605509
// MI455X (gfx1250) — compile-verified
//
#include <hip/hip_runtime.h>
#include <hip/hip_bf16.h>

#define NROWS 512
#define RANK  32
#define MCOLS 4096
#define PI_F  3.14159265358979f

typedef __attribute__((ext_vector_type(2))) float v2f;
typedef __attribute__((ext_vector_type(8))) float v8f;
typedef int b128_t __attribute__((vector_size(16)));   // matches builtin proto

// ---- CDNA5 async-to-LDS support (guarded; falls back to VGPR staging) -----
#if defined(__gfx1250__) && __has_builtin(__builtin_amdgcn_global_load_async_to_lds_b128)
#define HAVE_ASYNC_LDS 1
#else
#define HAVE_ASYNC_LDS 0
#endif

#if defined(__gfx1250__) && __has_builtin(__builtin_amdgcn_s_wait_asynccnt)
#define WAIT_ASYNC() __builtin_amdgcn_s_wait_asynccnt(0)
#elif defined(__HIP_DEVICE_COMPILE__)
#define WAIT_ASYNC() asm volatile("s_wait_asynccnt 0x0" ::: "memory")
#else
#define WAIT_ASYNC()
#endif

#define AS1 __attribute__((address_space(1)))
#define AS3 __attribute__((address_space(3)))

#if HAVE_ASYNC_LDS
static __device__ __forceinline__ void async_copy16(const void* g, void* l)
{
    // generic LDS pointers carry the LDS byte offset in their low 32 bits
    __builtin_amdgcn_global_load_async_to_lds_b128(
        (AS1 b128_t*)(unsigned long long)(size_t)g,
        (AS3 b128_t*)(unsigned)(size_t)l, 0, 0);
}
#endif

// stage one 32KB Hft row (2048 x 16B chunks) with 256 threads, 8 chunks each
static __device__ __forceinline__ void stage_row(float2* dst, const float2* src, int tid)
{
#if HAVE_ASYNC_LDS
    #pragma unroll
    for (int p = 0; p < 8; ++p) {
        const int c = tid + p * 256;                       // 16B units
        async_copy16((const char*)src + c * 16, (char*)dst + c * 16);
    }
#else
    #pragma unroll
    for (int p = 0; p < 8; ++p) {
        const int c = tid + p * 256;
        ((float4*)dst)[c] = ((const float4*)src)[c];
    }
#endif
}

// ---------------------------------------------------------------------------
// Kernel A: Hft[d,k] = sum_j H[d,j] * exp(-2*pi*i * j*k / M)
// Two f32 WMMA GEMMs (cos part, -sin part) against an LDS twiddle table
// gathered with running (j*k mod M) indices. 64 blocks x 8 waves; one 16x16
// output tile per wave; 1024 K-steps of V_WMMA_F32_16X16X4_F32 per GEMM.
// ---------------------------------------------------------------------------
__launch_bounds__(256)
__global__ void shiftnmf_dft_wmma(const float* __restrict__ H,
                                  float2* __restrict__ Hft)
{
    __shared__ float2 tw[MCOLS];              // 32 KB twiddle table in LDS
    const int tid = threadIdx.x;

    for (int t = tid; t < MCOLS; t += 256) {
        float s, c;
        float ang = (-2.0f * PI_F) * (float)t / (float)MCOLS;
        __sincosf(ang, &s, &c);               // exp(-i*2pi*t/M) = (c, s)
        tw[t] = make_float2(c, s);
    }
    __syncthreads();

    const int lane  = tid & 31;
    const int wave  = tid >> 5;
    const int tile  = blockIdx.x * 8 + wave;  // 0..511
    const int dtile = tile & 1;               // rank tile (2 x 16 rows)
    const int ftile = tile >> 1;              // freq tile (256 x 16 cols)

    // A fragment: lane holds row (lane&15), K = (lane>>4)*2 + {0,1}
    const int rowA  = dtile * 16 + (lane & 15);
    const int jlane = (lane >> 4) << 1;       // 0 or 2
    // B fragment: lane holds col (lane&15), same K split
    const int kf    = ftile * 16 + (lane & 15);

    const float* Arow = H + rowA * MCOLS + jlane;

    v8f cre = {0.f,0.f,0.f,0.f,0.f,0.f,0.f,0.f};
    v8f cim = {0.f,0.f,0.f,0.f,0.f,0.f,0.f,0.f};

    // running twiddle indices: idx_c(j0) = ((j0 + jlane + c) * kf) mod M
    unsigned idx0 = ((unsigned)(jlane + 0) * (unsigned)kf) & (MCOLS - 1);
    unsigned idx1 = ((unsigned)(jlane + 1) * (unsigned)kf) & (MCOLS - 1);
    const unsigned stepi = ((unsigned)kf << 2) & (MCOLS - 1);

    #pragma unroll 4
    for (int j0 = 0; j0 < MCOLS; j0 += 4) {
        __builtin_prefetch(Arow + j0 + 64, 0, 1);     // global_prefetch_b8
        const float2 av = *(const float2*)(Arow + j0);
        v2f a;  a.x = av.x;  a.y = av.y;

        const float2 t0 = tw[idx0];
        const float2 t1 = tw[idx1];
        v2f bc; bc.x = t0.x; bc.y = t1.x;             // cos part
        v2f bs; bs.x = t0.y; bs.y = t1.y;             // -sin part

        cre = __builtin_amdgcn_wmma_f32_16x16x4_f32(
                  false, a, false, bc, (short)0, cre, false, false);
        cim = __builtin_amdgcn_wmma_f32_16x16x4_f32(
                  false, a, false, bs, (short)0, cim, false, false);

        idx0 = (idx0 + stepi) & (MCOLS - 1);
        idx1 = (idx1 + stepi) & (MCOLS - 1);
    }

    // C/D layout: VGPR r -> row (lane>>4)*8 + r, col lane&15
    const int rbase = dtile * 16 + ((lane >> 4) << 3);
    #pragma unroll
    for (int r = 0; r < 8; ++r) {
        Hft[(rbase + r) * MCOLS + kf] = make_float2(cre[r], cim[r]);
    }
}

// ---------------------------------------------------------------------------
// Kernel B: V[n,k] = sum_d softplus(W[n,d]) * exp(-2*pi*i*tau[n,d]*k/M) * Hft[d,k]
// One workgroup per row n; 256 threads x 16 consecutive k each.
// Double-buffered async-to-LDS staging of the 32KB Hft rows (ASYNCcnt path):
// row d+1 DMA overlaps the phasor-rotation compute on row d.
// ---------------------------------------------------------------------------
__launch_bounds__(256)
__global__ void shiftnmf_mix(const float* __restrict__ W,
                             const float* __restrict__ tau,
                             const float2* __restrict__ Hft,
                             float2* __restrict__ V)
{
    __shared__ float2 rows[2][MCOLS];   // 2 x 32 KB double buffer
    __shared__ float  wsp_s[RANK];
    __shared__ float  tau_s[RANK];

    const int n   = blockIdx.x;
    const int tid = threadIdx.x;
    const int k0  = tid * 16;

    if (tid < RANK) {
        float x = W[n * RANK + tid];
        wsp_s[tid] = (x > 20.0f) ? x : log1pf(__expf(x));   // softplus
        tau_s[tid] = tau[n * RANK + tid];
    }

    float accRe[16], accIm[16];
    #pragma unroll
    for (int i = 0; i < 16; ++i) { accRe[i] = 0.0f; accIm[i] = 0.0f; }

    // prologue: stage row 0
    stage_row(rows[0], Hft, tid);
    WAIT_ASYNC();
    __syncthreads();

    for (int d = 0; d < RANK; ++d) {
        const float2* cur = rows[d & 1];

        // overlap: start DMA of row d+1 into the other buffer
        if (d + 1 < RANK)
            stage_row(rows[(d + 1) & 1], Hft + (d + 1) * MCOLS, tid);

        const float w  = wsp_s[d];
        const float th = (-2.0f * PI_F / (float)MCOLS) * tau_s[d];
        float s0, c0, ss, cs;
        __sincosf(th * (float)k0, &s0, &c0);                // phasor at k0
        __sincosf(th, &ss, &cs);                            // per-k rotation
        float cr = w * c0, ci = w * s0;                     // w * e^{i th k0}

        #pragma unroll
        for (int i = 0; i < 16; ++i) {
            const float2 h = cur[k0 + i];
            accRe[i] = fmaf(cr, h.x, fmaf(-ci, h.y, accRe[i]));
            accIm[i] = fmaf(cr, h.y, fmaf( ci, h.x, accIm[i]));
            const float nr = fmaf(cr, cs, -ci * ss);        // rotate phasor
            const float ni = fmaf(cr, ss,  ci * cs);
            cr = nr; ci = ni;
        }

        WAIT_ASYNC();       // my async copies of row d+1 landed in LDS
        __syncthreads();    // all waves done with cur + done staging
    }

    #pragma unroll
    for (int i = 0; i < 16; ++i) {
        V[n * MCOLS + k0 + i] = make_float2(accRe[i], accIm[i]);
    }
}

// ---------------------------------------------------------------------------
// inputs (setup_inputs order): 0=X (unused by the math), 1=W, 2=H, 3=tau
// output: V complex64 (512,4096) -> interleaved float pairs
// workspace: Hft (32 x 4096 complex) = 1 MB at d_ws
// ---------------------------------------------------------------------------
extern "C" void kernel_launch(void* const* d_in, const int* in_sizes, int n_in,
                              void* d_out, int out_size, void* d_ws, size_t ws_size,
                              hipStream_t stream)
{
    const float* W   = (const float*)d_in[1];
    const float* H   = (const float*)d_in[2];
    const float* tau = (const float*)d_in[3];
    float2* Hft = (float2*)d_ws;     // 1 MB scratch, fully overwritten each call
    float2* Vov = (float2*)d_out;

    shiftnmf_dft_wmma<<<64, 256, 0, stream>>>(H, Hft);
    shiftnmf_mix<<<NROWS, 256, 0, stream>>>(W, tau, Hft, Vov);
}